// PointPillarScatter_8753143349331
// MI455X (gfx1250) — compile-verified
//
#include <hip/hip_runtime.h>

// PointPillarScatter for MI455X (gfx1250).
// Zero-FLOP, write-bandwidth-bound scatter: no matrix work, so no WMMA.
// MI455X-specific path: b128 stores + TH=NT cache hints. The 446 MB output
// exceeds the 192 MB L2 and is never re-read on device -> stream it.
// Floor at 23.3 TB/s HBM: ~20-30 us, dominated by the zero-fill.

#define PP_NX 432
#define PP_NY 496
#define PP_C  64
#define PP_S  (PP_NX * PP_NY)   // 214272 cells per sample

// Clang vector types (NOT HIP_vector_type) so the nontemporal builtins
// accept them and lower to global_{load,store}_b128 th:*_NT.
typedef float v4f __attribute__((ext_vector_type(4)));
typedef int   v4i __attribute__((ext_vector_type(4)));

// ---------------------------------------------------------------------------
// Kernel 1: zero the entire output with 128-bit non-temporal stores.
// out_size floats, divisible by 4 (S*65*B is a multiple of 4).
// Emits: global_store_b128 v, v[..], s[..] scale_offset th:TH_STORE_NT
// ---------------------------------------------------------------------------
__global__ void pp_zero_fill_nt(v4f* __restrict__ out, unsigned n4) {
    unsigned i = blockIdx.x * blockDim.x + threadIdx.x;
    if (i < n4) {
        v4f z = {0.0f, 0.0f, 0.0f, 0.0f};
        __builtin_nontemporal_store(z, out + i);
    }
}

// ---------------------------------------------------------------------------
// Kernel 2: scatter. Thread t -> pillar p = t>>6, channel c = t&63.
// A wave32 covers 32 consecutive channels of one pillar: the feature read
// is one fully-coalesced 128 B line per wave (NT: each element read once).
// The coords row is one b128 load (reused by 64 threads -> keep cacheable).
// Writes are inherently random (unique (y,x) cells, stride S between channel
// planes) -> NT 4 B stores. Lane c==0 also scatters the point count.
// ---------------------------------------------------------------------------
__global__ void pp_scatter_nt(const float* __restrict__ feats,
                              const v4i*   __restrict__ coords,
                              const float* __restrict__ npts,
                              float*       __restrict__ out,
                              int P, int feat_total) {
    int gid = blockIdx.x * blockDim.x + threadIdx.x;
    int p = gid >> 6;
    int c = gid & 63;
    if (p >= P) return;

    // coords row: (batch, z, y, x); z is always 0 in this config but keep
    // the reference formula sidx = z + y*NX + x.
    v4i cr = coords[p];                              // single b128 load
    int b = cr.x;
    int sidx = cr.y + cr.z * PP_NX + cr.w;

    float v = __builtin_nontemporal_load(feats + (size_t)p * PP_C + c);
    size_t off = (size_t)(b * PP_C + c) * PP_S + (size_t)sidx;
    __builtin_nontemporal_store(v, out + off);       // global_store_b32 th:NT

    if (c == 0) {
        size_t poff = (size_t)feat_total + (size_t)b * PP_S + (size_t)sidx;
        __builtin_nontemporal_store(npts[p], out + poff);
    }
}

// ---------------------------------------------------------------------------
// Launch
//   d_in[0] pillar_features  [P, 64]  float32
//   d_in[1] voxel_coords     [P, 4]   int32 (b, z, y, x)
//   d_in[2] voxel_num_points [P]      float32
//   d_in[3] batch_size       [1]      (unused on device; derived from out_size)
//   d_out: features [B,64,NY,NX] then points [B,1,NY,NX], flat, float32
// ---------------------------------------------------------------------------
extern "C" void kernel_launch(void* const* d_in, const int* in_sizes, int n_in,
                              void* d_out, int out_size, void* d_ws, size_t ws_size,
                              hipStream_t stream) {
    (void)n_in; (void)d_ws; (void)ws_size;

    const float* feats  = (const float*)d_in[0];
    const v4i*   coords = (const v4i*)d_in[1];
    const float* npts   = (const float*)d_in[2];
    float*       out    = (float*)d_out;

    const int P = in_sizes[0] / PP_C;                 // 96000
    const int B = out_size / (PP_S * (PP_C + 1));     // 8
    const int feat_total = B * PP_C * PP_S;           // offset of points tensor

    // 1) stream-zero the whole output (446 MB of b128 NT stores)
    {
        unsigned n4 = (unsigned)(out_size / 4);
        unsigned threads = 256;                        // 8 waves per block
        unsigned blocks = (n4 + threads - 1) / threads;
        pp_zero_fill_nt<<<blocks, threads, 0, stream>>>((v4f*)out, n4);
    }

    // 2) scatter pillars (P*64 threads; 4 pillars per 256-thread block)
    {
        unsigned total = (unsigned)P * PP_C;
        unsigned threads = 256;
        unsigned blocks = (total + threads - 1) / threads;
        pp_scatter_nt<<<blocks, threads, 0, stream>>>(feats, coords, npts, out,
                                                      P, feat_total);
    }
}